// AlphaDetector_54322746360045
// MI455X (gfx1250) — compile-verified
//
#include <hip/hip_runtime.h>
#include <hip/hip_bf16.h>
#include <stdint.h>

typedef __attribute__((ext_vector_type(2))) float v2f;
typedef __attribute__((ext_vector_type(8))) float v8f;

namespace {
constexpr int   kBatch     = 32;
constexpr int   kH         = 384;
constexpr int   kW         = 384;
constexpr int   kDim       = 64;
constexpr int   kPat       = 512;              // 2^9 neighborhood patterns
constexpr int   kRowsPerWG = 8;                // output rows per block
constexpr int   kYBlocks   = kH / kRowsPerWG;  // 48
constexpr int   kRowDw     = kW / 32;          // 12 packed dwords per row
constexpr int   kLdsRows   = kRowsPerWG + 2;   // 10 (strip + halo)
constexpr int   kLdsDw     = kRowDw + 2;       // 14 (1-dword halo each side)
constexpr float kEps       = 1e-5f;
constexpr float kInvN      = 1.f / (float)(kH * kW);
// ws layout: [0, 16K) u32 hist[32][512]; then f32 table[512][64].
constexpr int   kHistElems  = kBatch * kPat;   // 16384 u32
constexpr int   kTableElems = kPat * kDim;     // 32768 f32
}

// ---- Kernel 0: zero the per-image pattern histograms (ws is poisoned). ----
__global__ __launch_bounds__(256)
void k0_zero_hist(uint32_t* __restrict__ hist) {
  const int i = blockIdx.x * 256 + threadIdx.x;
  if (i < kHistElems) hist[i] = 0u;
}

// ---- Kernel 1: per-image 3x3-neighborhood pattern histogram. ----
// Block = (image, 8-row strip), 256 threads (8 waves).
// Phase A: pack the strip's LSB bit-plane (rows y0-1..y0+8) into LDS bitmasks
//          via wave32 ballot (1 dword per 32 pixels, zero halo dword each side).
// Phase B: per pixel, extract the 3-bit window of 3 rows with a 64-bit shift,
//          form the 9-bit code k=dy*3+dx, ds_add_u32 into a 512-bin LDS hist.
// Phase C: merge to global with integer atomics (deterministic).
__global__ __launch_bounds__(256)
void k1_histogram(const float* __restrict__ alpha, uint32_t* __restrict__ hist) {
  const int b    = blockIdx.x / kYBlocks;
  const int yb   = blockIdx.x % kYBlocks;
  const int tid  = threadIdx.x;
  const int wave = tid >> 5;
  const int lane = tid & 31;
  const int y0   = yb * kRowsPerWG;
  const float* __restrict__ img = alpha + (size_t)b * kH * kW;

  __shared__ uint32_t bitrow[kLdsRows][kLdsDw];  // 140 dwords
  __shared__ uint32_t shist[kPat];

  // zero LDS (halo dwords + hist)
  for (int i = tid; i < kLdsRows * kLdsDw; i += 256) ((uint32_t*)bitrow)[i] = 0u;
  for (int i = tid; i < kPat; i += 256) shist[i] = 0u;

  // Phase A: 10 rows x 12 dword-groups = 120 ballot units.
  for (int u = wave; u < kLdsRows * kRowDw; u += 8) {
    const int r  = u / kRowDw;           // LDS row 0..9
    const int d  = u % kRowDw;           // dword group
    const int gy = y0 - 1 + r;           // global row in [-1, 384]
    const int gx = d * 32 + lane;
    int bit = 0;
    if (gy >= 0 && gy < kH) bit = ((int)(img[gy * kW + gx] * 255.f)) & 1;
    const uint32_t mask = __builtin_amdgcn_ballot_w32(bit != 0);
    if (lane == 0) bitrow[r][1 + d] = mask;
  }
  __syncthreads();

  // Phase B: 8 rows x 12 groups = 96 units; lane i handles pixel x = d*32+i.
  for (int u = wave; u < kRowsPerWG * kRowDw; u += 8) {
    const int rl = u / kRowDw + 1;       // LDS row of the output row
    const int d  = u % kRowDw;
    const int x  = d * 32 + lane;
    const int q  = x + 31;               // overall bit index of column x-1
    const int idx = q >> 5, sh = q & 31;
    uint32_t code = 0;
    #pragma unroll
    for (int dy = 0; dy < 3; ++dy) {
      const uint32_t lo = bitrow[rl - 1 + dy][idx];
      const uint32_t hi = bitrow[rl - 1 + dy][idx + 1];
      const uint32_t f  = (uint32_t)((((uint64_t)hi << 32) | lo) >> sh) & 7u;
      code |= f << (3 * dy);             // taps k = dy*3 + dx
    }
    atomicAdd(&shist[code], 1u);
  }
  __syncthreads();

  // Phase C: merge (integer atomics -> order-independent, deterministic).
  for (int e = tid; e < kPat; e += 256) {
    const uint32_t c = shist[e];
    if (c) atomicAdd(&hist[b * kPat + e], c);
  }
}

// ---- Kernel 2: pattern->channel value table, BN+bias+ReLU folded, fp32. ----
__global__ __launch_bounds__(256)
void k2_table(const float* __restrict__ conv_w, const float* __restrict__ conv_b,
              const float* __restrict__ bn_gamma, const float* __restrict__ bn_beta,
              const float* __restrict__ bn_mean, const float* __restrict__ bn_var,
              float* __restrict__ table) {
  const int i = blockIdx.x * 256 + threadIdx.x;
  if (i >= kTableElems) return;
  const int p = i >> 6;                  // pattern 0..511
  const int c = i & 63;                  // channel
  const float s = bn_gamma[c] * rsqrtf(bn_var[c] + kEps);
  float v = (conv_b[c] - bn_mean[c]) * s + bn_beta[c];
  #pragma unroll
  for (int k = 0; k < 9; ++k) {
    if ((p >> k) & 1) v += conv_w[c * 9 + k] * s;
  }
  table[p * kDim + c] = v > 0.f ? v : 0.f;   // ReLU baked in
}

// ---- Kernel 3: features = (1/N) * Hist[32x512] x Table[512x64] via ----
// V_WMMA_F32_16X16X4_F32 (exact fp32 matrix path; counts exact in f32).
// One workgroup, 8 waves: wave -> (m-tile of 16 images, n-tile of 16 ch),
// 128 chained k-steps (fixed order => deterministic). Plus the marker column.
__global__ __launch_bounds__(256)
void k3_gemm(const uint32_t* __restrict__ hist, const float* __restrict__ table,
             const float* __restrict__ alpha, float* __restrict__ out) {
  const int tid    = threadIdx.x;
  const int wave   = tid >> 5;
  const int lane   = tid & 31;
  const int lane16 = lane & 15;
  const int mt     = wave >> 2;          // 0..1  (images 16*mt..)
  const int nt     = wave & 3;           // 0..3  (channels 16*nt..)
  const int dk     = (lane >= 16) ? 2 : 0;
  const int img    = mt * 16 + lane16;   // A: M = lane%16
  const int ch     = nt * 16 + lane16;   // B/D: N = lane%16

  v8f acc = {};
  for (int kk = 0; kk < kPat / 4; ++kk) {
    const int k0 = kk * 4 + dk;
    // A 16x4 f32: VGPR0 = K=kk*4 (lanes<16) / K=kk*4+2 (lanes>=16); VGPR1 = +1.
    v2f a;
    a.x = (float)hist[img * kPat + k0];
    a.y = (float)hist[img * kPat + k0 + 1];
    // B 4x16 f32: rows striped the same way across the two half-waves.
    v2f bv;
    bv.x = table[(k0) * kDim + ch];
    bv.y = table[(k0 + 1) * kDim + ch];
    acc = __builtin_amdgcn_wmma_f32_16x16x4_f32(false, a, false, bv,
                                                (short)0, acc, false, false);
  }
  // D 16x16 f32: lane holds channel N=ch for images M = j + 8*(lane>=16).
  #pragma unroll
  for (int j = 0; j < 8; ++j) {
    const int m = mt * 16 + j + ((lane >= 16) ? 8 : 0);
    out[m * (kDim + 1) + ch] = acc[j] * kInvN;
  }
  if (tid == 0) {
    const int marker[4] = {65, 73, 50, 52};   // 'A' 'I' '2' '4'
    int ok = 1;
    for (int j = 0; j < 4; ++j) {
      int by = 0;
      for (int i = 0; i < 8; ++i) {
        const int bit = ((int)(alpha[j * 8 + i] * 255.f)) & 1;
        by |= bit << (7 - i);                 // MSB-first packing
      }
      if (by != marker[j]) ok = 0;
    }
    const float mv = ok ? 1.f : 0.f;
    for (int b = 0; b < kBatch; ++b) out[b * (kDim + 1) + kDim] = mv;
  }
}

extern "C" void kernel_launch(void* const* d_in, const int* in_sizes, int n_in,
                              void* d_out, int out_size, void* d_ws, size_t ws_size,
                              hipStream_t stream) {
  (void)in_sizes; (void)n_in; (void)out_size; (void)ws_size;
  const float* alpha    = (const float*)d_in[0];
  const float* conv_w   = (const float*)d_in[1];
  const float* conv_b   = (const float*)d_in[2];
  const float* bn_gamma = (const float*)d_in[3];
  const float* bn_beta  = (const float*)d_in[4];
  const float* bn_mean  = (const float*)d_in[5];
  const float* bn_var   = (const float*)d_in[6];
  float* out = (float*)d_out;

  uint32_t* hist  = (uint32_t*)d_ws;                      // 64 KB
  float*    table = (float*)((char*)d_ws + kHistElems * sizeof(uint32_t)); // 128 KB

  k0_zero_hist<<<dim3((kHistElems + 255) / 256), dim3(256), 0, stream>>>(hist);
  k1_histogram<<<dim3(kBatch * kYBlocks), dim3(256), 0, stream>>>(alpha, hist);
  k2_table<<<dim3((kTableElems + 255) / 256), dim3(256), 0, stream>>>(
      conv_w, conv_b, bn_gamma, bn_beta, bn_mean, bn_var, table);
  k3_gemm<<<dim3(1), dim3(256), 0, stream>>>(hist, table, alpha, out);
}